// LSSBased2DTO3D_62405874811760
// MI455X (gfx1250) — compile-verified
//
#include <hip/hip_runtime.h>

typedef __attribute__((ext_vector_type(2))) float v2f;
typedef __attribute__((ext_vector_type(8))) float v8f;

#define B_   2
#define V_   6
#define D_   48
#define H_   16
#define W_   44
#define C_   128
#define GX   128
#define GY   128
#define ODIM 256
#define M_TOTAL (B_ * GY * GX)   // 32768 BEV rows

// ---------------------------------------------------------------------------
// Kernel 1: zero the BEV accumulator (B * GY*GX * C floats = 16 MB)
// ---------------------------------------------------------------------------
__global__ void zero_bev_kernel(float4* __restrict__ bev) {
    size_t i = (size_t)blockIdx.x * blockDim.x + threadIdx.x;
    bev[i] = make_float4(0.f, 0.f, 0.f, 0.f);
}

// ---------------------------------------------------------------------------
// Kernel 2: depth softmax + weighted scatter into BEV grid.
// One block per (B, v, h, w) pixel; 128 threads = one per channel c.
// ---------------------------------------------------------------------------
__global__ void scatter_kernel(const float* __restrict__ frustum,  // (B,v,c,h,w)
                               const float* __restrict__ coords,   // (B,v,d,h,w,2)
                               const float* __restrict__ logits,   // (B,v,d,h,w)
                               float* __restrict__ bev)            // (B*GY*GX, C)
{
    __shared__ float wgt[D_];
    __shared__ int   seg[D_];
    __shared__ float inv_sum;

    int pix = blockIdx.x;                 // B*v*h*w pixels
    int wi  = pix % W_;  int tmp = pix / W_;
    int hi  = tmp % H_;  tmp /= H_;
    int vi  = tmp % V_;
    int bi  = tmp / V_;

    const int tid = threadIdx.x;          // 0..127 == channel index

    const size_t dstride = (size_t)H_ * W_;
    const size_t base0 = (((size_t)bi * V_ + vi) * D_) * dstride + (size_t)hi * W_ + wi;

    if (tid < D_) {
        size_t idx = base0 + (size_t)tid * dstride;
        wgt[tid] = logits[idx];
        float cx = coords[idx * 2 + 0];
        float cy = coords[idx * 2 + 1];
        int x = (int)floorf((cx + 1.0f) * 0.5f * (float)GX);
        int y = (int)floorf((cy + 1.0f) * 0.5f * (float)GY);
        bool valid = (x >= 0) && (x < GX) && (y >= 0) && (y < GY);
        seg[tid] = valid ? (bi * (GY * GX) + y * GX + x) : -1;
    }
    __syncthreads();

    if (tid == 0) {
        float m = wgt[0];
        #pragma unroll
        for (int i = 1; i < D_; ++i) m = fmaxf(m, wgt[i]);
        float s = 0.f;
        for (int i = 0; i < D_; ++i) { float e = expf(wgt[i] - m); wgt[i] = e; s += e; }
        inv_sum = 1.0f / s;
    }
    __syncthreads();

    const float inv = inv_sum;
    const float f = frustum[((((size_t)bi * V_ + vi) * C_ + tid) * H_ + hi) * W_ + wi];

    for (int di = 0; di < D_; ++di) {
        int s = seg[di];
        if (s >= 0) {
            atomicAdd(&bev[(size_t)s * C_ + tid], wgt[di] * inv * f);
        }
    }
}

// ---------------------------------------------------------------------------
// Kernel 3: out = bev(32768x128) @ W(128x256) + bias, full f32 via
// V_WMMA_F32_16X16X4_F32. 8 waves/block; one 16x16 C tile per wave.
// A frag (16x4 f32): lane holds M=lane&15, K-pair (lane>>4)*2  -> 2 VGPRs
// B frag (4x16 f32): lane holds N=lane&15, K-pair (lane>>4)*2  -> 2 VGPRs
// C/D  : VGPR r holds (M = r + (lane>>4)*8, N = lane&15)
// ---------------------------------------------------------------------------
__global__ void gemm_wmma_kernel(const float* __restrict__ A,     // (M_TOTAL, C_)
                                 const float* __restrict__ Wm,    // (C_, ODIM)
                                 const float* __restrict__ bias,  // (ODIM)
                                 float* __restrict__ out)         // (M_TOTAL, ODIM)
{
    const int lane = threadIdx.x & 31;
    const int wave = threadIdx.x >> 5;
    const int tile = blockIdx.x * 8 + wave;
    const int nTile = tile % (ODIM / 16);
    const int mTile = tile / (ODIM / 16);
    const int mBase = mTile * 16;
    const int nBase = nTile * 16;

    const int rowm  = lane & 15;         // A row / B,C column within tile
    const int khalf = (lane >> 4) * 2;   // 0 for lanes 0-15, 2 for lanes 16-31

    v8f acc = {};
    const float* Arow = A + (size_t)(mBase + rowm) * C_ + khalf;   // 8B aligned
    const float* Bcol = Wm + nBase + rowm;

    #pragma unroll
    for (int k0 = 0; k0 < C_; k0 += 4) {
        v2f a = *(const v2f*)(Arow + k0);                       // A[row][k0+khalf .. +1]
        v2f b;
        b.x = Bcol[(size_t)(k0 + khalf + 0) * ODIM];            // W[k][col]
        b.y = Bcol[(size_t)(k0 + khalf + 1) * ODIM];
        acc = __builtin_amdgcn_wmma_f32_16x16x4_f32(
                  /*neg_a=*/false, a, /*neg_b=*/false, b,
                  /*c_mod=*/(short)0, acc, /*reuse_a=*/false, /*reuse_b=*/false);
    }

    const int mOff = (lane >> 4) * 8;
    const int ncol = lane & 15;
    const float bv = bias[nBase + ncol];
    #pragma unroll
    for (int r = 0; r < 8; ++r) {
        out[(size_t)(mBase + mOff + r) * ODIM + nBase + ncol] = acc[r] + bv;
    }
}

// ---------------------------------------------------------------------------
extern "C" void kernel_launch(void* const* d_in, const int* in_sizes, int n_in,
                              void* d_out, int out_size, void* d_ws, size_t ws_size,
                              hipStream_t stream) {
    // inputs per setup_inputs() order:
    // 0: bev_features (unused by reference math, shape only)
    // 1: frustum_features (B,v,c,h,w)
    // 2: frustum_bev_coordinates (B,v,d,h,w,2)
    // 3: last_step_depth_features (B,v,d,h,w)
    // 4: W_align (c, out_dim)
    // 5: b_align (out_dim)
    const float* frustum = (const float*)d_in[1];
    const float* coords  = (const float*)d_in[2];
    const float* logits  = (const float*)d_in[3];
    const float* Wm      = (const float*)d_in[4];
    const float* bias    = (const float*)d_in[5];
    float* out = (float*)d_out;
    float* bev = (float*)d_ws;   // M_TOTAL * C_ floats = 16 MB scratch

    // 1) zero accumulator: M_TOTAL*C_ = 4,194,304 floats = 1,048,576 float4
    zero_bev_kernel<<<(M_TOTAL * C_ / 4) / 256, 256, 0, stream>>>((float4*)bev);

    // 2) softmax + scatter: one block per pixel
    scatter_kernel<<<B_ * V_ * H_ * W_, 128, 0, stream>>>(frustum, coords, logits, bev);

    // 3) WMMA GEMM + bias: 32768 tiles of 16x16, 8 waves per block
    const int tiles = (M_TOTAL / 16) * (ODIM / 16);
    gemm_wmma_kernel<<<tiles / 8, 256, 0, stream>>>(bev, Wm, bias, out);
}